// DifferentiableNeuralMemory_29317446762720
// MI455X (gfx1250) — compile-verified
//
#include <hip/hip_runtime.h>

// ---------------------------------------------------------------------------
// Differentiable Neural Memory on gfx1250 (MI455X).
//  - Sequential scan: 4 kernels/step x 2048 steps, all state resident in L2.
//  - Final batched read: two f32 GEMMs via V_WMMA_F32_16X16X4_F32, 16x64
//    macro-tile per wave (4 accumulators) with explicit register
//    double-buffering so B-fragment loads overlap the WMMA chain.
// ---------------------------------------------------------------------------

#define DIM     1024
#define TSTEPS  2048
#define BQ      1024
#define AHID    32
#define DECAYF  0.9f
#define KEEPF   0.99f      // (1 - FORGET)
#define MAXGRAD 10.0f
#define MAXMOM  1.0f
#define MAXW    5.0f

typedef __attribute__((ext_vector_type(2))) float v2f;
typedef __attribute__((ext_vector_type(8))) float v8f;

__device__ __forceinline__ float sigmoidf_(float x) { return 1.0f / (1.0f + __expf(-x)); }
__device__ __forceinline__ float siluf_(float x)    { return x * sigmoidf_(x); }
__device__ __forceinline__ float clampf_(float x, float lo, float hi) {
  return fminf(fmaxf(x, lo), hi);
}

// ---- init: copy W0/W1/b0/b1 into mutable workspace, zero momenta ----------
__global__ void k_init(const float* __restrict__ W0, const float* __restrict__ W1,
                       const float* __restrict__ b0, const float* __restrict__ b1,
                       float* __restrict__ W0c, float* __restrict__ W1c,
                       float* __restrict__ mW0, float* __restrict__ mW1,
                       float* __restrict__ b0c, float* __restrict__ b1c,
                       float* __restrict__ mb0, float* __restrict__ mb1) {
  size_t i = (size_t)blockIdx.x * blockDim.x + threadIdx.x;
  size_t n = (size_t)DIM * DIM;
  if (i < n)   { W0c[i] = W0[i]; W1c[i] = W1[i]; mW0[i] = 0.f; mW1[i] = 0.f; }
  if (i < DIM) { b0c[i] = b0[i]; b1c[i] = b1[i]; mb0[i] = 0.f; mb1[i] = 0.f; }
}

// ---- stage 1: z0 = W0c @ k + b0 ; h1 = silu(z0). wave-per-row -------------
__global__ void k_fwd0(const float* __restrict__ W0c, const float* __restrict__ b0c,
                       const float* __restrict__ kv,
                       float* __restrict__ z0, float* __restrict__ h1,
                       float* __restrict__ scal) {
  if (blockIdx.x == 0 && threadIdx.x == 0) scal[0] = 0.0f;  // reset ||g||^2 accum
  int row  = blockIdx.x * (blockDim.x >> 5) + (threadIdx.x >> 5);
  int lane = threadIdx.x & 31;
  const float* wr = W0c + (size_t)row * DIM;
  float acc = 0.0f;
  for (int c = lane; c < DIM; c += 32) acc += wr[c] * kv[c];
  for (int o = 16; o > 0; o >>= 1) acc += __shfl_down(acc, o, 32);
  if (lane == 0) {
    float z = acc + b0c[row];
    z0[row] = z;
    h1[row] = siluf_(z);
  }
}

// ---- stage 2: pred = W1c @ h1 + b1 ; g_raw = 2(pred - v); accumulate ||g||^2
__global__ void k_fwd1(const float* __restrict__ W1c, const float* __restrict__ b1c,
                       const float* __restrict__ h1, const float* __restrict__ vv,
                       float* __restrict__ graw, float* __restrict__ scal) {
  int row  = blockIdx.x * (blockDim.x >> 5) + (threadIdx.x >> 5);
  int lane = threadIdx.x & 31;
  const float* wr = W1c + (size_t)row * DIM;
  float acc = 0.0f;
  for (int c = lane; c < DIM; c += 32) acc += wr[c] * h1[c];
  for (int o = 16; o > 0; o >>= 1) acc += __shfl_down(acc, o, 32);
  if (lane == 0) {
    float g = 2.0f * (acc + b1c[row] - vv[row]);
    graw[row] = g;
    atomicAdd(&scal[0], g * g);
  }
}

// ---- stage 3: gh1 = W1ᵀ g, gz0 = gh1 * silu'(z0). thread-per-column -------
// Block 0's first wave also evaluates the alpha MLP (H=32, one lane each).
__global__ void k_bwd(const float* __restrict__ W1c, const float* __restrict__ graw,
                      const float* __restrict__ z0, float* __restrict__ gz0,
                      const float* __restrict__ aw1, const float* __restrict__ ab1,
                      const float* __restrict__ aw2, const float* __restrict__ ab2,
                      const float* __restrict__ alpha_scale,
                      float* __restrict__ scal) {
  float gn = sqrtf(scal[0]);
  float gs = fminf(1.0f, MAXGRAD / gn);            // grad-norm clip factor
  if (blockIdx.x == 0 && threadIdx.x < 32) {
    int h = threadIdx.x;
    float surprise = gn * gs;                       // ||g|| after clipping
    float ahv = siluf_(aw1[h] * surprise + ab1[h]);
    float p = aw2[h] * ahv;
    for (int o = 16; o > 0; o >>= 1) p += __shfl_down(p, o, 32);
    if (h == 0) {
      scal[1] = gs;
      scal[2] = sigmoidf_(p + ab2[0]) * alpha_scale[0];
    }
  }
  int j = blockIdx.x * blockDim.x + threadIdx.x;    // column of W1
  float acc = 0.0f;
#pragma unroll 8
  for (int r = 0; r < DIM; ++r) acc += W1c[(size_t)r * DIM + j] * graw[r];
  acc *= gs;                                        // gh1[j] with clipped g
  float z = z0[j];
  float s = sigmoidf_(z);
  gz0[j] = acc * (s * (1.0f + z * (1.0f - s)));     // silu'(z0)
}

// ---- stage 4: fused rank-1 momentum + forget + write + clamp --------------
__global__ void k_upd(float* __restrict__ W0c, float* __restrict__ W1c,
                      float* __restrict__ mW0, float* __restrict__ mW1,
                      float* __restrict__ b0c, float* __restrict__ b1c,
                      float* __restrict__ mb0, float* __restrict__ mb1,
                      const float* __restrict__ graw, const float* __restrict__ gz0,
                      const float* __restrict__ h1, const float* __restrict__ kv,
                      const float* __restrict__ scal) {
  size_t idx = (size_t)blockIdx.x * blockDim.x + threadIdx.x;
  float gs = scal[1], alpha = scal[2];
  int i = (int)(idx >> 10);          // row (uniform within a block)
  int j = (int)(idx & (DIM - 1));    // col (coalesced)
  float gi = graw[i] * gs;           // g[i]
  float m1 = clampf_(DECAYF * mW1[idx] - alpha * (gi * h1[j]), -MAXMOM, MAXMOM);
  mW1[idx] = m1;
  W1c[idx] = clampf_(KEEPF * W1c[idx] + m1, -MAXW, MAXW);
  float gz = gz0[i];
  float m0 = clampf_(DECAYF * mW0[idx] - alpha * (gz * kv[j]), -MAXMOM, MAXMOM);
  mW0[idx] = m0;
  W0c[idx] = clampf_(KEEPF * W0c[idx] + m0, -MAXW, MAXW);
  if (idx < DIM) {                   // bias updates (not clamped after write)
    float n1 = clampf_(DECAYF * mb1[idx] - alpha * (graw[idx] * gs), -MAXMOM, MAXMOM);
    mb1[idx] = n1; b1c[idx] = KEEPF * b1c[idx] + n1;
    float n0 = clampf_(DECAYF * mb0[idx] - alpha * gz0[idx], -MAXMOM, MAXMOM);
    mb0[idx] = n0; b0c[idx] = KEEPF * b0c[idx] + n0;
  }
}

// ---- final batched read: C = act(A @ Wᵀ + bias) via f32 WMMA --------------
// One wave owns a 16x64 macro-tile of C (4 independent accumulators).
// Explicit register double-buffering: K-step n+1 fragments (1xA + 4xB, each
// a single 8B-aligned b64 load) are issued BEFORE the K-step n WMMA quartet,
// so the waitcnt pass can overlap loads with the WMMA chain instead of
// draining loadcnt to 0 in front of every WMMA.
// A tile 16x4 per-lane layout: lane(half,l15) -> (M=l15, K=half*2 .. +1).
// B tile 4x16: B[k][n] = W[n][k]; lane(half,l15) -> (K=half*2..+1, N=l15).
__global__ void k_gemm_wmma(const float* __restrict__ A, const float* __restrict__ W,
                            const float* __restrict__ bias, float* __restrict__ C,
                            int apply_silu) {
  int waveId = blockIdx.x * (blockDim.x >> 5) + (threadIdx.x >> 5);
  int lane   = threadIdx.x & 31;
  int tilesN = DIM / 64;                       // 16 macro-columns
  int tm = (waveId / tilesN) * 16;
  int tn = (waveId % tilesN) * 64;
  int half = lane >> 4;                        // 0 or 1
  int l15  = lane & 15;
  const float* arow  = A + (size_t)(tm + l15) * DIM + half * 2;
  const float* brow0 = W + (size_t)(tn +  0 + l15) * DIM + half * 2;
  const float* brow1 = W + (size_t)(tn + 16 + l15) * DIM + half * 2;
  const float* brow2 = W + (size_t)(tn + 32 + l15) * DIM + half * 2;
  const float* brow3 = W + (size_t)(tn + 48 + l15) * DIM + half * 2;

  v8f c0 = {}, c1 = {}, c2 = {}, c3 = {};
  // Prologue: load K-step 0 fragments.
  v2f a  = *(const v2f*)(arow);
  v2f b0 = *(const v2f*)(brow0);
  v2f b1 = *(const v2f*)(brow1);
  v2f b2 = *(const v2f*)(brow2);
  v2f b3 = *(const v2f*)(brow3);

#pragma unroll 4
  for (int d = 0; d < DIM - 4; d += 4) {
    // Issue next-step loads first (stay outstanding across the WMMA quartet).
    v2f an  = *(const v2f*)(arow  + d + 4);
    v2f bn0 = *(const v2f*)(brow0 + d + 4);
    v2f bn1 = *(const v2f*)(brow1 + d + 4);
    v2f bn2 = *(const v2f*)(brow2 + d + 4);
    v2f bn3 = *(const v2f*)(brow3 + d + 4);
    // (neg_a, A, neg_b, B, c_mod, C, reuse_a, reuse_b)
    c0 = __builtin_amdgcn_wmma_f32_16x16x4_f32(false, a, false, b0, (short)0, c0, false, false);
    c1 = __builtin_amdgcn_wmma_f32_16x16x4_f32(false, a, false, b1, (short)0, c1, false, false);
    c2 = __builtin_amdgcn_wmma_f32_16x16x4_f32(false, a, false, b2, (short)0, c2, false, false);
    c3 = __builtin_amdgcn_wmma_f32_16x16x4_f32(false, a, false, b3, (short)0, c3, false, false);
    a = an; b0 = bn0; b1 = bn1; b2 = bn2; b3 = bn3;
  }
  // Epilogue K-step.
  c0 = __builtin_amdgcn_wmma_f32_16x16x4_f32(false, a, false, b0, (short)0, c0, false, false);
  c1 = __builtin_amdgcn_wmma_f32_16x16x4_f32(false, a, false, b1, (short)0, c1, false, false);
  c2 = __builtin_amdgcn_wmma_f32_16x16x4_f32(false, a, false, b2, (short)0, c2, false, false);
  c3 = __builtin_amdgcn_wmma_f32_16x16x4_f32(false, a, false, b3, (short)0, c3, false, false);

  // Store: VGPR r of each accumulator holds M=r (lanes 0-15), M=r+8 (16-31).
#pragma unroll
  for (int r = 0; r < 8; ++r) {
    int m = tm + r + half * 8;
    float v0 = c0[r] + bias[tn +  0 + l15];
    float v1 = c1[r] + bias[tn + 16 + l15];
    float v2 = c2[r] + bias[tn + 32 + l15];
    float v3 = c3[r] + bias[tn + 48 + l15];
    if (apply_silu) { v0 = siluf_(v0); v1 = siluf_(v1); v2 = siluf_(v2); v3 = siluf_(v3); }
    C[(size_t)m * DIM + tn +  0 + l15] = v0;
    C[(size_t)m * DIM + tn + 16 + l15] = v1;
    C[(size_t)m * DIM + tn + 32 + l15] = v2;
    C[(size_t)m * DIM + tn + 48 + l15] = v3;
  }
}

// ---------------------------------------------------------------------------
extern "C" void kernel_launch(void* const* d_in, const int* in_sizes, int n_in,
                              void* d_out, int out_size, void* d_ws, size_t ws_size,
                              hipStream_t stream) {
  const float* keys        = (const float*)d_in[0];
  const float* values      = (const float*)d_in[1];
  const float* queries     = (const float*)d_in[2];
  const float* W0          = (const float*)d_in[3];
  const float* W1          = (const float*)d_in[4];
  const float* b0          = (const float*)d_in[5];
  const float* b1          = (const float*)d_in[6];
  const float* aw1         = (const float*)d_in[7];
  const float* ab1         = (const float*)d_in[8];
  const float* aw2         = (const float*)d_in[9];
  const float* ab2         = (const float*)d_in[10];
  const float* alpha_scale = (const float*)d_in[11];
  float* out = (float*)d_out;

  // Workspace partition (floats): 4 DxD state matrices, vectors, scalars, C1.
  float* ws  = (float*)d_ws;
  size_t DD  = (size_t)DIM * DIM;
  float* W0c = ws;
  float* W1c = W0c + DD;
  float* mW0 = W1c + DD;
  float* mW1 = mW0 + DD;
  float* b0c = mW1 + DD;
  float* b1c = b0c + DIM;
  float* mb0 = b1c + DIM;
  float* mb1 = mb0 + DIM;
  float* z0v = mb1 + DIM;
  float* h1v = z0v + DIM;
  float* grw = h1v + DIM;
  float* gzv = grw + DIM;
  float* scl = gzv + DIM;      // [0]=||g_raw||^2, [1]=clip scale, [2]=alpha
  float* C1  = scl + 16;       // (B x D) hidden activations of final read

  k_init<<<(unsigned)(DD / 256), 256, 0, stream>>>(W0, W1, b0, b1,
                                                   W0c, W1c, mW0, mW1,
                                                   b0c, b1c, mb0, mb1);

  // Sequential surprise writes: 4 stream-ordered kernels per step.
  for (int t = 0; t < TSTEPS; ++t) {
    const float* kv = keys   + (size_t)t * DIM;
    const float* vv = values + (size_t)t * DIM;
    k_fwd0<<<DIM / 8, 256, 0, stream>>>(W0c, b0c, kv, z0v, h1v, scl);
    k_fwd1<<<DIM / 8, 256, 0, stream>>>(W1c, b1c, h1v, vv, grw, scl);
    k_bwd <<<DIM / 256, 256, 0, stream>>>(W1c, grw, z0v, gzv,
                                          aw1, ab1, aw2, ab2, alpha_scale, scl);
    k_upd <<<(unsigned)(DD / 256), 256, 0, stream>>>(W0c, W1c, mW0, mW1,
                                                     b0c, b1c, mb0, mb1,
                                                     grw, gzv, h1v, kv, scl);
  }

  // Batched read through the final memory state: two WMMA GEMMs.
  // macro-tiles = (BQ/16)*(DIM/64) = 1024 waves; 8 waves (256 thr) per block.
  unsigned gemmBlocks = (BQ / 16) * (DIM / 64) / 8;
  k_gemm_wmma<<<gemmBlocks, 256, 0, stream>>>(queries, W0c, b0c, C1, 1);
  k_gemm_wmma<<<gemmBlocks, 256, 0, stream>>>(C1,      W1c, b1c, out, 0);
}